// ScaledDotProductTAMCaD_31353261260858
// MI455X (gfx1250) — compile-verified
//
#include <hip/hip_runtime.h>

// Temporal attention (TAMCaD): B=4, G=64, H=64, T=2048, 4 heads, dk=16.
// One block = (b, 4 consecutive t). 16 waves: wave w -> head (w&3), t-local (w>>2).
// fp32 WMMA 16x16x4 for both matmuls (memory-bound problem -> keep full precision).

#define TDIM  2048
#define TT    4
#define BLOCK 512   // 16 wave32

typedef float v2f __attribute__((ext_vector_type(2)));
typedef float v8f __attribute__((ext_vector_type(8)));

// LDS float-offsets (total 81920 floats = 320 KB, the full WGP budget)
#define QS_OFF 0        // q staged [hid][tl][g]   : 64*4*64
#define KS_OFF 16384    // k staged [hid][tl][g]
#define VS_OFF 32768    // v staged [g][tl][hid]
#define AP_OFF 0        // per-wave attn, transposed [g'][g] (aliases staging after s2)
#define MB_OFF 65536    // head-mean buffer / x buffer [tl][g][col] : 4*64*64
#define LDS_BYTES (81920 * 4)

__global__ __launch_bounds__(BLOCK, 1)
void tamcad_attn_kernel(const float* __restrict__ q,
                        const float* __restrict__ k,
                        const float* __restrict__ v,
                        float* __restrict__ out_x,
                        float* __restrict__ out_attn) {
  extern __shared__ float lds[];
  const int tid  = threadIdx.x;
  const int w    = tid >> 5;
  const int lane = tid & 31;
  const int lidx = lane & 15;    // position within 16-lane half
  const int hf   = lane >> 4;    // which half (K-pair select for WMMA frags)
  const int h    = w & 3;        // head
  const int tl   = w >> 2;       // local t (0..3)
  const int b    = blockIdx.y;
  const int t0   = blockIdx.x * TT;
  const size_t baseB = (size_t)b * 4096 * TDIM;

  // zero the head-mean accumulator
  for (int i = tid; i < 16384; i += BLOCK) lds[MB_OFF + i] = 0.0f;

  // ---- stage q,k,v: float4 along t, scatter into transpose-friendly LDS ----
#pragma unroll
  for (int it = 0; it < 8; ++it) {
    const int r = tid + it * BLOCK;
    {
      const int hid = r >> 6, g = r & 63;   // consecutive tid -> consecutive g (no bank conflict)
      const float4 vq = *(const float4*)(q + baseB + (size_t)(g * 64 + hid) * TDIM + t0);
      lds[QS_OFF + (hid * 4 + 0) * 64 + g] = vq.x;
      lds[QS_OFF + (hid * 4 + 1) * 64 + g] = vq.y;
      lds[QS_OFF + (hid * 4 + 2) * 64 + g] = vq.z;
      lds[QS_OFF + (hid * 4 + 3) * 64 + g] = vq.w;
      const float4 vk = *(const float4*)(k + baseB + (size_t)(g * 64 + hid) * TDIM + t0);
      lds[KS_OFF + (hid * 4 + 0) * 64 + g] = vk.x;
      lds[KS_OFF + (hid * 4 + 1) * 64 + g] = vk.y;
      lds[KS_OFF + (hid * 4 + 2) * 64 + g] = vk.z;
      lds[KS_OFF + (hid * 4 + 3) * 64 + g] = vk.w;
    }
    {
      const int g = r >> 6, hid = r & 63;   // consecutive tid -> consecutive hid
      const float4 vv = *(const float4*)(v + baseB + (size_t)r * TDIM + t0);
      lds[VS_OFF + (g * 4 + 0) * 64 + hid] = vv.x;
      lds[VS_OFF + (g * 4 + 1) * 64 + hid] = vv.y;
      lds[VS_OFF + (g * 4 + 2) * 64 + hid] = vv.z;
      lds[VS_OFF + (g * 4 + 3) * 64 + hid] = vv.w;
    }
  }
  __syncthreads();  // s1: staging complete

  // ---- logits = q(64x16) @ k^T(16x64) via 16 C-tiles, K in 4 chunks of 4 ----
  v8f acc[4][4];
#pragma unroll
  for (int mt = 0; mt < 4; ++mt)
#pragma unroll
    for (int nt = 0; nt < 4; ++nt)
#pragma unroll
      for (int e = 0; e < 8; ++e) acc[mt][nt][e] = 0.0f;

#pragma unroll
  for (int kc = 0; kc < 4; ++kc) {
    const int hb = h * 16 + kc * 4 + hf * 2;   // hidden index of K-pair base
    v2f qf[4], kf[4];
#pragma unroll
    for (int mt = 0; mt < 4; ++mt) {
      const int g = mt * 16 + lidx;
      qf[mt].x = lds[QS_OFF + ((hb + 0) * 4 + tl) * 64 + g];
      qf[mt].y = lds[QS_OFF + ((hb + 1) * 4 + tl) * 64 + g];
    }
#pragma unroll
    for (int nt = 0; nt < 4; ++nt) {
      const int gp = nt * 16 + lidx;
      kf[nt].x = lds[KS_OFF + ((hb + 0) * 4 + tl) * 64 + gp];
      kf[nt].y = lds[KS_OFF + ((hb + 1) * 4 + tl) * 64 + gp];
    }
#pragma unroll
    for (int mt = 0; mt < 4; ++mt)
#pragma unroll
      for (int nt = 0; nt < 4; ++nt)
        acc[mt][nt] = __builtin_amdgcn_wmma_f32_16x16x4_f32(
            false, qf[mt], false, kf[nt], (short)0, acc[mt][nt], false, false);
  }

  // ---- softmax over g' (row lives in one VGPR across a 16-lane half) ----
  const float scale = 0.25f;  // dk^-0.5
#pragma unroll
  for (int mt = 0; mt < 4; ++mt) {
#pragma unroll
    for (int r = 0; r < 8; ++r) {
      float m = fmaxf(fmaxf(acc[mt][0][r], acc[mt][1][r]),
                      fmaxf(acc[mt][2][r], acc[mt][3][r]));
      m = fmaxf(m, __shfl_xor(m, 1));
      m = fmaxf(m, __shfl_xor(m, 2));
      m = fmaxf(m, __shfl_xor(m, 4));
      m = fmaxf(m, __shfl_xor(m, 8));
      float s = 0.0f;
#pragma unroll
      for (int nt = 0; nt < 4; ++nt) {
        const float p = __expf((acc[mt][nt][r] - m) * scale);
        acc[mt][nt][r] = p;
        s += p;
      }
      s += __shfl_xor(s, 1);
      s += __shfl_xor(s, 2);
      s += __shfl_xor(s, 4);
      s += __shfl_xor(s, 8);
      const float rinv = 1.0f / s;
#pragma unroll
      for (int nt = 0; nt < 4; ++nt) acc[mt][nt][r] *= rinv;
    }
  }

  // ---- pull v B-fragments into registers before staging LDS is recycled ----
  v2f vf[16];
#pragma unroll
  for (int kc2 = 0; kc2 < 16; ++kc2) {
    const int gp = kc2 * 4 + hf * 2;
    vf[kc2].x = lds[VS_OFF + ((gp + 0) * 4 + tl) * 64 + h * 16 + lidx];
    vf[kc2].y = lds[VS_OFF + ((gp + 1) * 4 + tl) * 64 + h * 16 + lidx];
  }
  __syncthreads();  // s2: all staging reads done; safe to overwrite with attn

  // ---- attn -> per-wave LDS (transposed [g'][g]) + head-mean accumulate ----
  float* ap = lds + AP_OFF + w * 4096;
  float* mb = lds + MB_OFF;
#pragma unroll
  for (int mt = 0; mt < 4; ++mt)
#pragma unroll
    for (int nt = 0; nt < 4; ++nt)
#pragma unroll
      for (int r = 0; r < 8; ++r) {
        const int g  = mt * 16 + r + hf * 8;
        const int gp = nt * 16 + lidx;
        const float val = acc[mt][nt][r];
        ap[gp * 64 + g] = val;
        atomicAdd(&mb[(tl * 64 + g) * 64 + gp], val * 0.25f);  // ds_add_f32
      }

  // ---- x = attn(64x64) @ v(64x16): A-frags from transposed LDS (conflict-free) ----
  v8f xacc[4];
#pragma unroll
  for (int mt = 0; mt < 4; ++mt)
#pragma unroll
    for (int e = 0; e < 8; ++e) xacc[mt][e] = 0.0f;

#pragma unroll
  for (int kc2 = 0; kc2 < 16; ++kc2) {
    const int gpb = kc2 * 4 + hf * 2;
#pragma unroll
    for (int mt = 0; mt < 4; ++mt) {
      v2f a;
      a.x = ap[(gpb + 0) * 64 + mt * 16 + lidx];
      a.y = ap[(gpb + 1) * 64 + mt * 16 + lidx];
      xacc[mt] = __builtin_amdgcn_wmma_f32_16x16x4_f32(
          false, a, false, vf[kc2], (short)0, xacc[mt], false, false);
    }
  }
  __syncthreads();  // s3: mean adds complete

  // ---- cooperative coalesced store of attn head-mean (float4 along t) ----
#pragma unroll
  for (int it = 0; it < 8; ++it) {
    const int rr = tid + it * BLOCK;   // rr = g*64 + g'
    float4 o;
    o.x = mb[rr];
    o.y = mb[4096 + rr];
    o.z = mb[8192 + rr];
    o.w = mb[12288 + rr];
    *(float4*)(out_attn + baseB + (size_t)rr * TDIM + t0) = o;
  }
  __syncthreads();  // s4: mean buffer free, reuse for x

  // ---- x tiles -> shared buffer [tl][g][h*16+d] ----
#pragma unroll
  for (int mt = 0; mt < 4; ++mt)
#pragma unroll
    for (int r = 0; r < 8; ++r) {
      const int g = mt * 16 + r + hf * 8;
      mb[(tl * 64 + g) * 64 + h * 16 + lidx] = xacc[mt][r];
    }
  __syncthreads();  // s5

  // ---- cooperative coalesced store of x (float4 along t) ----
#pragma unroll
  for (int it = 0; it < 8; ++it) {
    const int rr = tid + it * BLOCK;   // rr = g*64 + h*16 + d
    float4 o;
    o.x = mb[rr];
    o.y = mb[4096 + rr];
    o.z = mb[8192 + rr];
    o.w = mb[12288 + rr];
    *(float4*)(out_x + baseB + (size_t)rr * TDIM + t0) = o;
  }
}

extern "C" void kernel_launch(void* const* d_in, const int* in_sizes, int n_in,
                              void* d_out, int out_size, void* d_ws, size_t ws_size,
                              hipStream_t stream) {
  const float* q = (const float*)d_in[0];
  const float* k = (const float*)d_in[1];
  const float* v = (const float*)d_in[2];
  float* out_x    = (float*)d_out;
  float* out_attn = (float*)d_out + (size_t)4 * 4096 * TDIM;  // x first, then attn

  dim3 grid(TDIM / TT, 4);  // (t-tiles, batch)
  tamcad_attn_kernel<<<grid, BLOCK, LDS_BYTES, stream>>>(q, k, v, out_x, out_attn);
}